// GATConvGlobal_56951266345487
// MI455X (gfx1250) — compile-verified
//
#include <hip/hip_runtime.h>
#include <hip/hip_bf16.h>

typedef __attribute__((ext_vector_type(2))) float v2f;
typedef __attribute__((ext_vector_type(8))) float v8f;

#define INF_ 128
#define OUTW 256        // OUT*H
#define LSTRIDE 264     // padded LDS row stride (floats): 264 % 64 == 8 -> conflict-free
#define SLOPE 0.2f

__device__ __forceinline__ float lrelu(float x) { return x >= 0.f ? x : SLOPE * x; }

// order-preserving float <-> uint key for atomic max
__device__ __forceinline__ unsigned fkey(float f) {
    unsigned u = __float_as_uint(f);
    return (u & 0x80000000u) ? ~u : (u | 0x80000000u);
}
__device__ __forceinline__ float fdecode(unsigned k) {
    unsigned u = (k & 0x80000000u) ? (k & 0x7FFFFFFFu) : ~k;
    return __uint_as_float(u);
}

// ---------------------------------------------------------------------------
// Fused GEMM (X[N,128] @ W[128,256]) + per-row attention dots.
// One wave per 16-row strip; 16 N-tiles of V_WMMA_F32_16X16X4_F32 each.
// W staged through LDS (bank-conflict-free padded rows) in two 64-row K
// phases, using gfx1250 async memory->LDS copies (ASYNCcnt-tracked).
// ---------------------------------------------------------------------------
__global__ __launch_bounds__(256) void gemm_attn(
    const float* __restrict__ X, const float* __restrict__ W,
    const float* __restrict__ attn_l, const float* __restrict__ attn_r,
    float* __restrict__ Y, float* __restrict__ el, float* __restrict__ er,
    int N)
{
    extern __shared__ float Wlds[];  // 64 * LSTRIDE floats (= 66KB)

    const int wave = threadIdx.x >> 5;
    const int lane = threadIdx.x & 31;
    const int half = lane >> 4;      // 0: lanes 0-15, 1: lanes 16-31
    const int l15  = lane & 15;
    const int strip = blockIdx.x * 8 + wave;
    const int row0  = strip * 16;
    const bool valid = (row0 < N);

    int rowA = row0 + l15;
    if (rowA >= N) rowA = N - 1;               // clamp; stores are guarded
    const float* xrow = X + (size_t)rowA * INF_;

    // LDS byte address of the staging buffer (generic-pointer low 32 bits)
    const unsigned ldsBase = (unsigned)(uintptr_t)Wlds;

    const v8f vzero = {0.f,0.f,0.f,0.f,0.f,0.f,0.f,0.f};
    v8f acc[16];
#pragma unroll
    for (int t = 0; t < 16; ++t) acc[t] = vzero;

    for (int ph = 0; ph < 2; ++ph) {
        // ---- stage 64 rows of W into LDS via async memory->LDS copies ----
        // 64 rows x 256 floats = 4096 float4 chunks; 16 chunks per thread.
        const char* gbase = (const char*)(W + (size_t)ph * 64 * OUTW);
#pragma unroll 1
        for (int i = threadIdx.x; i < 64 * (OUTW / 4); i += 256) {
            const int row    = i >> 6;          // 64 float4 per source row
            const int within = i & 63;
            const void* gaddr = gbase + (size_t)row * (OUTW * 4) + (size_t)within * 16;
            const unsigned laddr = ldsBase + (unsigned)(row * (LSTRIDE * 4) + within * 16);
            asm volatile("global_load_async_to_lds_b128 %0, %1, off"
                         :: "v"(laddr), "v"(gaddr)
                         : "memory");
        }
        asm volatile("s_wait_asynccnt 0x0" ::: "memory");
        __syncthreads();

        if (valid) {
#pragma unroll 1
            for (int kk = 0; kk < 16; ++kk) {
                const int kg = ph * 64 + kk * 4;
                // A fragment (16x4 f32): VGPR0 = K0(lanes<16)/K2, VGPR1 = K1/K3
                v2f a;
                a.x = xrow[kg + half * 2 + 0];
                a.y = xrow[kg + half * 2 + 1];
#pragma unroll
                for (int t = 0; t < 16; ++t) {
                    const int col = t * 16 + l15;
                    v2f b;
                    b.x = Wlds[(kk * 4 + half * 2 + 0) * LSTRIDE + col];
                    b.y = Wlds[(kk * 4 + half * 2 + 1) * LSTRIDE + col];
                    acc[t] = __builtin_amdgcn_wmma_f32_16x16x4_f32(
                        false, a, false, b, (short)0, acc[t], false, false);
                }
            }
        }
        __syncthreads();
    }

    if (!valid) return;

    // epilogue: store Y tiles + fused attention dot products
    float elp[2][8], erp[2][8];
#pragma unroll
    for (int hd = 0; hd < 2; ++hd)
#pragma unroll
        for (int i = 0; i < 8; ++i) { elp[hd][i] = 0.f; erp[hd][i] = 0.f; }

#pragma unroll
    for (int t = 0; t < 16; ++t) {
        const int hd = t >> 3;                     // head = col/128
        const int d  = (t & 7) * 16 + l15;         // within-head dim
        const float al = attn_l[hd * 128 + d];
        const float ar = attn_r[hd * 128 + d];
#pragma unroll
        for (int i = 0; i < 8; ++i) {
            const float v = acc[t][i];
            const int m = row0 + i + half * 8;     // C/D layout: M = i (+8 upper half)
            if (m < N) Y[(size_t)m * OUTW + t * 16 + l15] = v;
            elp[hd][i] += v * al;
            erp[hd][i] += v * ar;
        }
    }

    // reduce across the 16 lanes of each half (columns of one row)
#pragma unroll
    for (int hd = 0; hd < 2; ++hd) {
#pragma unroll
        for (int i = 0; i < 8; ++i) {
            float sl = elp[hd][i], sr = erp[hd][i];
#pragma unroll
            for (int m = 8; m >= 1; m >>= 1) {
                sl += __shfl_xor(sl, m, 16);
                sr += __shfl_xor(sr, m, 16);
            }
            if (l15 == 0) {
                const int row = row0 + i + half * 8;
                if (row < N) {
                    el[(size_t)row * 2 + hd] = sl;
                    er[(size_t)row * 2 + hd] = sr;
                }
            }
        }
    }
}

// ---------------------------------------------------------------------------
__global__ void zero_f32(float* p, long long n) {
    long long i = (long long)blockIdx.x * blockDim.x + threadIdx.x;
    if (i < n) p[i] = 0.f;
}

// per-bond raw logits + segment max (encoded-uint atomicMax)
__global__ void edge_logits(const float* __restrict__ el_h, const float* __restrict__ er_a,
                            const float* __restrict__ el_e,
                            const int* __restrict__ src, const int* __restrict__ dst,
                            float* __restrict__ eh, float* __restrict__ ee,
                            unsigned* __restrict__ emaxkey, long long n2)
{
    long long idx = (long long)blockIdx.x * blockDim.x + threadIdx.x;
    if (idx >= n2) return;
    const long long b = idx >> 1;
    const int hd = (int)(idx & 1);
    const int s = src[b], d = dst[b];
    const float rv = er_a[(size_t)d * 2 + hd];
    const float vh = lrelu(el_h[(size_t)s * 2 + hd] + rv);
    const float ve = lrelu(el_e[idx] + rv);
    eh[idx] = vh;
    ee[idx] = ve;
    atomicMax(&emaxkey[(size_t)d * 2 + hd], fkey(fmaxf(vh, ve)));
}

// per-atom: global->atom logit, final max, exp, init esum
__global__ void atom_eu(const float* __restrict__ el_u, const float* __restrict__ er_a,
                        const int* __restrict__ gid, const unsigned* __restrict__ emaxkey,
                        float* __restrict__ emax, float* __restrict__ eu,
                        float* __restrict__ esum, long long n2)
{
    long long idx = (long long)blockIdx.x * blockDim.x + threadIdx.x;
    if (idx >= n2) return;
    const long long a = idx >> 1;
    const int hd = (int)(idx & 1);
    const float raw = lrelu(el_u[(size_t)gid[a] * 2 + hd] + er_a[idx]);
    const float mb = fdecode(emaxkey[idx]);   // -NaN when segment empty
    const float m = fmaxf(raw, mb);           // fmaxf drops the NaN
    emax[idx] = m;
    const float ex = expf(raw - m);
    eu[idx] = ex;
    esum[idx] = ex;
}

// per-bond: exponentiate logits, accumulate softmax denominator
__global__ void edge_exp(const float* __restrict__ emax, const int* __restrict__ dst,
                         float* __restrict__ eh, float* __restrict__ ee,
                         float* __restrict__ esum, long long n2)
{
    long long idx = (long long)blockIdx.x * blockDim.x + threadIdx.x;
    if (idx >= n2) return;
    const long long b = idx >> 1;
    const int hd = (int)(idx & 1);
    const int d = dst[b];
    const float m = emax[(size_t)d * 2 + hd];
    const float a = expf(eh[idx] - m);
    const float c = expf(ee[idx] - m);
    eh[idx] = a;
    ee[idx] = c;
    atomicAdd(&esum[(size_t)d * 2 + hd], a + c);
}

// per-bond weighted scatter-add: out[dst] += h[src]*alpha_h + e*alpha_e
// one block per bond, 256 lanes = full feature width (out lives in L2: 102MB)
__global__ __launch_bounds__(256) void edge_agg(
    const float* __restrict__ h, const float* __restrict__ e,
    const float* __restrict__ eh, const float* __restrict__ ee,
    const float* __restrict__ esum,
    const int* __restrict__ src, const int* __restrict__ dst,
    float* __restrict__ outagg)
{
    const long long b = blockIdx.x;
    const int c = threadIdx.x;
    const int hd = c >> 7;
    const int s = src[b], d = dst[b];
    const float inv = 1.0f / esum[(size_t)d * 2 + hd];
    const float ah = eh[b * 2 + hd] * inv;
    const float ae = ee[b * 2 + hd] * inv;
    const float v = h[(size_t)s * OUTW + c] * ah + e[(size_t)b * OUTW + c] * ae;
    atomicAdd(&outagg[(size_t)d * OUTW + c], v);
}

// head mean (+ global-feature term) + BN batch statistics partials
__global__ __launch_bounds__(128) void mean_stats(
    const float* __restrict__ outagg, const float* __restrict__ u,
    const int* __restrict__ gid,
    const float* __restrict__ eu, const float* __restrict__ esum,
    float* __restrict__ hm, float* __restrict__ colsum, float* __restrict__ colsq,
    int Na)
{
    const int d = threadIdx.x;
    float s = 0.f, s2 = 0.f;
    for (int a = blockIdx.x; a < Na; a += gridDim.x) {
        const int g = gid[a];
        const float a0 = eu[(size_t)a * 2 + 0] / esum[(size_t)a * 2 + 0];
        const float a1 = eu[(size_t)a * 2 + 1] / esum[(size_t)a * 2 + 1];
        const float v0 = outagg[(size_t)a * OUTW + d]       + u[(size_t)g * OUTW + d]       * a0;
        const float v1 = outagg[(size_t)a * OUTW + 128 + d] + u[(size_t)g * OUTW + 128 + d] * a1;
        const float m = 0.5f * (v0 + v1);
        hm[(size_t)a * 128 + d] = m;
        s += m;
        s2 += m * m;
    }
    atomicAdd(&colsum[d], s);
    atomicAdd(&colsq[d], s2);
}

// BatchNorm (batch stats) + ReLU
__global__ void bn_relu(const float* __restrict__ hm,
                        const float* __restrict__ colsum, const float* __restrict__ colsq,
                        const float* __restrict__ gamma, const float* __restrict__ beta,
                        float* __restrict__ out, int Na)
{
    long long idx = (long long)blockIdx.x * blockDim.x + threadIdx.x;
    if (idx >= (long long)Na * 128) return;
    const int d = (int)(idx & 127);
    const float invn = 1.0f / (float)Na;
    const float mean = colsum[d] * invn;
    const float var  = colsq[d] * invn - mean * mean;
    const float y = (hm[idx] - mean) * rsqrtf(var + 1e-5f) * gamma[d] + beta[d];
    out[idx] = fmaxf(y, 0.f);
}

// ---------------------------------------------------------------------------
extern "C" void kernel_launch(void* const* d_in, const int* in_sizes, int n_in,
                              void* d_out, int out_size, void* d_ws, size_t ws_size,
                              hipStream_t stream) {
    const float* atom   = (const float*)d_in[0];
    const float* bond   = (const float*)d_in[1];
    const float* glob   = (const float*)d_in[2];
    const float* Wa     = (const float*)d_in[3];
    const float* Wb     = (const float*)d_in[4];
    const float* Wg     = (const float*)d_in[5];
    const float* attn_l = (const float*)d_in[6];
    const float* attn_r = (const float*)d_in[7];
    const float* gamma  = (const float*)d_in[8];
    const float* beta   = (const float*)d_in[9];
    const int* src = (const int*)d_in[10];
    const int* dst = (const int*)d_in[11];
    const int* gid = (const int*)d_in[12];

    const int Na = in_sizes[0] / INF_;
    const int Nb = in_sizes[1] / INF_;
    const int Ng = in_sizes[2] / INF_;

    float* ws = (float*)d_ws;
    size_t off = 0;
    auto alloc = [&](size_t n) { float* p = ws + off; off += (n + 3) & ~(size_t)3; return p; };
    float* h     = alloc((size_t)Na * OUTW);
    float* e     = alloc((size_t)Nb * OUTW);
    float* u     = alloc((size_t)Ng * OUTW);
    float* el_h  = alloc((size_t)Na * 2);
    float* er_a  = alloc((size_t)Na * 2);
    float* el_e  = alloc((size_t)Nb * 2);
    float* er_b  = alloc((size_t)Nb * 2);   // unused by later stages
    float* el_u  = alloc((size_t)Ng * 2);
    float* er_g  = alloc((size_t)Ng * 2);   // unused by later stages
    float* eh    = alloc((size_t)Nb * 2);
    float* ee    = alloc((size_t)Nb * 2);
    unsigned* emaxkey = (unsigned*)alloc((size_t)Na * 2);
    float* emax  = alloc((size_t)Na * 2);
    float* eu    = alloc((size_t)Na * 2);
    float* esum  = alloc((size_t)Na * 2);
    float* outagg = alloc((size_t)Na * OUTW);
    float* colsum = alloc(128);
    float* colsq  = alloc(128);   // contiguous with colsum
    float* hm = h;                // h is dead after edge_agg; reuse for hm

    // init scratch that is accumulated into
    {
        long long nz = (long long)Na * 2;
        zero_f32<<<(int)((nz + 255) / 256), 256, 0, stream>>>((float*)emaxkey, nz);
        nz = (long long)Na * OUTW;
        zero_f32<<<(int)((nz + 255) / 256), 256, 0, stream>>>(outagg, nz);
        zero_f32<<<1, 256, 0, stream>>>(colsum, 256);  // colsum + colsq
    }

    const int ldsBytes = 64 * LSTRIDE * 4;  // 66KB padded double-phase weight stage
    auto nblk = [](int n) { int strips = (n + 15) / 16; return (strips + 7) / 8; };
    gemm_attn<<<nblk(Na), 256, ldsBytes, stream>>>(atom, Wa, attn_l, attn_r, h, el_h, er_a, Na);
    gemm_attn<<<nblk(Nb), 256, ldsBytes, stream>>>(bond, Wb, attn_l, attn_r, e, el_e, er_b, Nb);
    gemm_attn<<<nblk(Ng), 256, ldsBytes, stream>>>(glob, Wg, attn_l, attn_r, u, el_u, er_g, Ng);

    const long long n2b = (long long)Nb * 2;
    const long long n2a = (long long)Na * 2;
    edge_logits<<<(int)((n2b + 255) / 256), 256, 0, stream>>>(el_h, er_a, el_e, src, dst,
                                                              eh, ee, emaxkey, n2b);
    atom_eu<<<(int)((n2a + 255) / 256), 256, 0, stream>>>(el_u, er_a, gid, emaxkey,
                                                          emax, eu, esum, n2a);
    edge_exp<<<(int)((n2b + 255) / 256), 256, 0, stream>>>(emax, dst, eh, ee, esum, n2b);
    edge_agg<<<Nb, 256, 0, stream>>>(h, e, eh, ee, esum, src, dst, outagg);
    mean_stats<<<1024, 128, 0, stream>>>(outagg, u, gid, eu, esum, hm, colsum, colsq, Na);
    const long long nout = (long long)Na * 128;
    bn_relu<<<(int)((nout + 255) / 256), 256, 0, stream>>>(hm, colsum, colsq, gamma, beta,
                                                           (float*)d_out, Na);
}